// DeconvWithPruning_80418967650325
// MI455X (gfx1250) — compile-verified
//
#include <hip/hip_runtime.h>
#include <hip/hip_bf16.h>
#include <cstdint>
#include <cstddef>

typedef __attribute__((ext_vector_type(2))) float v2f;
typedef __attribute__((ext_vector_type(8))) float v8f;

#define NKERN 27
#define CIN   64
#define COUT  64
// 24-bit hash space (step=256, batch=0): 2^24 bits -> 2^19 32-bit words
#define NWORDS     (1u << 19)
#define TILE_WORDS 1024u
#define NTILES     (NWORDS / TILE_WORDS)   // 512

// ---------------------------------------------------------------------------
// helpers
// ---------------------------------------------------------------------------
__device__ __forceinline__ uint32_t dwp_cand_hash(const int* xc, int n, int k) {
  int x = xc[n * 4 + 1] + (k / 9) - 1;
  int y = xc[n * 4 + 2] + ((k / 3) % 3) - 1;
  int z = xc[n * 4 + 3] + (k % 3) - 1;
  return ((uint32_t)x << 16) | ((uint32_t)y << 8) | (uint32_t)z;
}

// ---------------------------------------------------------------------------
// 0) zero helpers (128-bit stores for the big region)
// ---------------------------------------------------------------------------
__global__ void dwp_zero128(float4* p, size_t n4) {
  size_t t = (size_t)blockIdx.x * blockDim.x + threadIdx.x;
  if (t < n4) p[t] = make_float4(0.f, 0.f, 0.f, 0.f);
}

// ---------------------------------------------------------------------------
// 1) mark candidate-hash presence bits
// ---------------------------------------------------------------------------
__global__ void dwp_mark_cand(const int* __restrict__ xc, uint32_t* candMask, int M) {
  int t = blockIdx.x * blockDim.x + threadIdx.x;
  if (t >= M) return;
  int n = t / NKERN, k = t % NKERN;
  uint32_t h = dwp_cand_hash(xc, n, k);
  atomicOr(&candMask[h >> 5], 1u << (h & 31u));
}

// ---------------------------------------------------------------------------
// 2) mark reference-hash presence bits
// ---------------------------------------------------------------------------
__global__ void dwp_mark_ref(const int* __restrict__ rc, uint32_t* refMask, int nref) {
  int t = blockIdx.x * blockDim.x + threadIdx.x;
  if (t >= nref) return;
  uint32_t h = ((uint32_t)rc[t * 4 + 1] << 16) |
               ((uint32_t)rc[t * 4 + 2] << 8)  |
               (uint32_t)rc[t * 4 + 3];
  atomicOr(&refMask[h >> 5], 1u << (h & 31u));
}

// ---------------------------------------------------------------------------
// 3) per-tile popcount prefix (exclusive within tile) + tile totals
// ---------------------------------------------------------------------------
__global__ void dwp_scan_tiles(const uint32_t* __restrict__ candMask,
                               uint32_t* __restrict__ wordPrefix,
                               uint32_t* __restrict__ tileSums) {
  __shared__ uint32_t sm[256];
  int tile = blockIdx.x;
  int t = threadIdx.x;  // 256 threads, 4 words each
  uint32_t base = tile * TILE_WORDS + (uint32_t)t * 4u;
  uint32_t c0 = (uint32_t)__builtin_popcount(candMask[base + 0]);
  uint32_t c1 = (uint32_t)__builtin_popcount(candMask[base + 1]);
  uint32_t c2 = (uint32_t)__builtin_popcount(candMask[base + 2]);
  uint32_t c3 = (uint32_t)__builtin_popcount(candMask[base + 3]);
  uint32_t s = c0 + c1 + c2 + c3;
  sm[t] = s;
  __syncthreads();
  for (int off = 1; off < 256; off <<= 1) {
    uint32_t v = (t >= off) ? sm[t - off] : 0u;
    __syncthreads();
    sm[t] += v;
    __syncthreads();
  }
  uint32_t run = sm[t] - s;  // exclusive prefix for this thread's 4 words
  wordPrefix[base + 0] = run;            run += c0;
  wordPrefix[base + 1] = run;            run += c1;
  wordPrefix[base + 2] = run;            run += c2;
  wordPrefix[base + 3] = run;
  if (t == 255) tileSums[tile] = sm[255];
}

// ---------------------------------------------------------------------------
// 4) exclusive scan of the 512 tile totals (single block)
// ---------------------------------------------------------------------------
__global__ void dwp_scan_tilesums(uint32_t* tileSums) {
  __shared__ uint32_t sm[NTILES];
  int t = threadIdx.x;
  uint32_t v = tileSums[t];
  sm[t] = v;
  __syncthreads();
  for (int off = 1; off < (int)NTILES; off <<= 1) {
    uint32_t u = (t >= off) ? sm[t - off] : 0u;
    __syncthreads();
    sm[t] += u;
    __syncthreads();
  }
  tileSums[t] = sm[t] - v;  // exclusive
}

// ---------------------------------------------------------------------------
// 5) fold tile offsets into per-word prefixes
// ---------------------------------------------------------------------------
__global__ void dwp_add_offsets(uint32_t* wordPrefix, const uint32_t* tileSums) {
  uint32_t w = blockIdx.x * blockDim.x + threadIdx.x;
  if (w < NWORDS) wordPrefix[w] += tileSums[w / TILE_WORDS];
}

// ---------------------------------------------------------------------------
// 6) default-fill all output rows: coords = cand[0] (= x_coords[0] + (-1,-1,-1)),
//    keep = 0  (rows beyond the unique count stay like this, matching the
//    JAX `unique(size=M)` padding where padded indices are 0)
// ---------------------------------------------------------------------------
__global__ void dwp_fill_defaults(const int* __restrict__ xc,
                                  float* __restrict__ outCoords,
                                  float* __restrict__ outKeep, int M) {
  int r = blockIdx.x * blockDim.x + threadIdx.x;
  if (r >= M) return;
  outCoords[(size_t)r * 4 + 0] = 0.0f;
  outCoords[(size_t)r * 4 + 1] = (float)(xc[1] - 1);
  outCoords[(size_t)r * 4 + 2] = (float)(xc[2] - 1);
  outCoords[(size_t)r * 4 + 3] = (float)(xc[3] - 1);
  outKeep[r] = 0.0f;
}

// ---------------------------------------------------------------------------
// 7) for every present hash (ascending), write decoded coords + keep flag
// ---------------------------------------------------------------------------
__global__ void dwp_write_unique(const uint32_t* __restrict__ candMask,
                                 const uint32_t* __restrict__ wordPrefix,
                                 const uint32_t* __restrict__ refMask,
                                 float* __restrict__ outCoords,
                                 float* __restrict__ outKeep) {
  uint32_t w = blockIdx.x * blockDim.x + threadIdx.x;
  if (w >= NWORDS) return;
  uint32_t m = candMask[w];
  uint32_t rank = wordPrefix[w];
  while (m) {
    int b = __builtin_ctz(m);
    m &= m - 1u;
    uint32_t h = (w << 5) | (uint32_t)b;
    outCoords[(size_t)rank * 4 + 0] = 0.0f;
    outCoords[(size_t)rank * 4 + 1] = (float)(h >> 16);
    outCoords[(size_t)rank * 4 + 2] = (float)((h >> 8) & 255u);
    outCoords[(size_t)rank * 4 + 3] = (float)(h & 255u);
    outKeep[rank] = (float)((refMask[h >> 5] >> (h & 31u)) & 1u);
    ++rank;
  }
}

// ---------------------------------------------------------------------------
// 8) fused GEMM (f32 WMMA 16x16x4) + pruned dedup scatter.
//    One block = one 16-row tile of x_feats; A fragments register-resident
//    across all 27 kernel offsets. 4 waves = 4 column tiles of C_out=64.
//    A whole offset is skipped (wave-uniform) when none of the 16 target
//    rows survives pruning; individual atomics are keep-predicated.
// ---------------------------------------------------------------------------
__global__ void __launch_bounds__(128) dwp_gemm_scatter(
    const float* __restrict__ xf, const int* __restrict__ xc,
    const float* __restrict__ W,
    const uint32_t* __restrict__ candMask,
    const uint32_t* __restrict__ wordPrefix,
    const uint32_t* __restrict__ refMask,
    float* __restrict__ outFeats) {
  int rowTile = blockIdx.x;
  int wid  = threadIdx.x >> 5;   // 0..3 : column tile
  int lane = threadIdx.x & 31;
  int hl   = lane >> 4;          // half-wave: K pair (A/B), M range (C/D)
  int lm   = lane & 15;          // M (A rows) / N (B,D cols)
  int rowBase = rowTile * 16;
  int colBase = wid * 16;

  // A fragments for the whole K=64 strip, loaded once (16 x v2f = 32 VGPRs)
  v2f afrag[CIN / 4];
  {
    const float* ap = xf + (size_t)(rowBase + lm) * CIN + 2 * hl;
#pragma unroll
    for (int s = 0; s < CIN / 4; ++s) {
      afrag[s].x = ap[4 * s + 0];
      afrag[s].y = ap[4 * s + 1];
    }
  }
  // this lane's row coordinates (rows duplicated across the two half-waves)
  int n_row = rowBase + lm;
  int cx = xc[n_row * 4 + 1];
  int cy = xc[n_row * 4 + 2];
  int cz = xc[n_row * 4 + 3];

#pragma unroll 1
  for (int koff = 0; koff < NKERN; ++koff) {
    int x = cx + (koff / 9) - 1;
    int y = cy + ((koff / 3) % 3) - 1;
    int z = cz + (koff % 3) - 1;
    uint32_t h = ((uint32_t)x << 16) | ((uint32_t)y << 8) | (uint32_t)z;
    uint32_t rbit = (refMask[h >> 5] >> (h & 31u)) & 1u;
    // wave-uniform early-out: every lane's target row is pruned away
    if (!__any((int)rbit)) continue;

    uint32_t seg = wordPrefix[h >> 5] +
                   (uint32_t)__builtin_popcount(candMask[h >> 5] &
                                                ((1u << (h & 31u)) - 1u));
    int packed = (int)((seg << 1) | rbit);

    const float* Wk = W + (size_t)koff * CIN * COUT + (size_t)(2 * hl) * COUT +
                      colBase + lm;
    v8f acc = {};
#pragma unroll
    for (int s = 0; s < CIN / 4; ++s) {
      v2f b;
      b.x = Wk[(size_t)(4 * s) * COUT];
      b.y = Wk[(size_t)(4 * s) * COUT + COUT];
      // D(16x16,f32) = A(16x4,f32) * B(4x16,f32) + C
      acc = __builtin_amdgcn_wmma_f32_16x16x4_f32(
          /*neg_a=*/false, afrag[s], /*neg_b=*/false, b,
          /*c_mod=*/(short)0, acc, /*reuse_a=*/false, /*reuse_b=*/false);
    }
#pragma unroll
    for (int i = 0; i < 8; ++i) {
      int row = i + 8 * hl;  // D VGPR i holds matrix row i + 8*(lane>>4)
      int pk = __shfl(packed, row, 32);
      if (pk & 1)
        atomicAdd(&outFeats[(size_t)(pk >> 1) * COUT + colBase + lm], acc[i]);
    }
  }
}

// scalar tail (rows not covered by full 16-row tiles; zero work when N%16==0)
__global__ void dwp_gemm_tail(const float* __restrict__ xf, const int* __restrict__ xc,
                              const float* __restrict__ W,
                              const uint32_t* __restrict__ candMask,
                              const uint32_t* __restrict__ wordPrefix,
                              const uint32_t* __restrict__ refMask,
                              float* __restrict__ outFeats, int rowStart, int rem) {
  int t = blockIdx.x * blockDim.x + threadIdx.x;
  if (t >= rem * NKERN * COUT) return;
  int c    = t % COUT;
  int koff = (t / COUT) % NKERN;
  int r    = rowStart + t / (COUT * NKERN);
  uint32_t h = dwp_cand_hash(xc, r, koff);
  if (!((refMask[h >> 5] >> (h & 31u)) & 1u)) return;  // pruned target row
  const float* Wk = W + (size_t)koff * CIN * COUT;
  float s = 0.0f;
  for (int k = 0; k < CIN; ++k) s += xf[(size_t)r * CIN + k] * Wk[(size_t)k * COUT + c];
  uint32_t seg = wordPrefix[h >> 5] +
                 (uint32_t)__builtin_popcount(candMask[h >> 5] & ((1u << (h & 31u)) - 1u));
  atomicAdd(&outFeats[(size_t)seg * COUT + c], s);
}

// ---------------------------------------------------------------------------
// 9) finalize: kept rows get +bias; non-kept rows are untouched (already zero)
// ---------------------------------------------------------------------------
__global__ void dwp_finalize(float* __restrict__ outFeats,
                             const float* __restrict__ outKeep,
                             const float* __restrict__ bias, size_t total) {
  size_t t = (size_t)blockIdx.x * blockDim.x + threadIdx.x;
  if (t >= total) return;
  size_t r = t >> 6;
  if (outKeep[r] > 0.5f) outFeats[t] += bias[(int)(t & 63u)];
}

// ---------------------------------------------------------------------------
extern "C" void kernel_launch(void* const* d_in, const int* in_sizes, int n_in,
                              void* d_out, int out_size, void* d_ws, size_t ws_size,
                              hipStream_t stream) {
  const float* xf   = (const float*)d_in[0];  // [N,64]
  const int*   xc   = (const int*)d_in[1];    // [N,4]
  const int*   rc   = (const int*)d_in[2];    // [NREF,4]
  const float* W    = (const float*)d_in[3];  // [27,64,64]
  const float* bias = (const float*)d_in[4];  // [64]

  const int N    = in_sizes[0] / CIN;
  const int NREF = in_sizes[2] / 4;
  const int M    = N * NKERN;

  // workspace layout (u32): candMask | refMask | wordPrefix | tileSums
  uint32_t* candMask   = (uint32_t*)d_ws;
  uint32_t* refMask    = candMask + NWORDS;
  uint32_t* wordPrefix = refMask + NWORDS;
  uint32_t* tileSums   = wordPrefix + NWORDS;

  // output layout (float): coords [M,4] | feats [M,64] | keep [M]
  float* outCoords = (float*)d_out;
  float* outFeats  = outCoords + (size_t)M * 4;
  float* outKeep   = outFeats + (size_t)M * COUT;

  const int TB = 256;
  // 0) zero the two bitmasks and the accumulation region (128-bit stores)
  {
    size_t nm4 = (2u * NWORDS) / 4;
    dwp_zero128<<<(unsigned)((nm4 + TB - 1) / TB), TB, 0, stream>>>((float4*)candMask, nm4);
    size_t nf4 = ((size_t)M * COUT) / 4;
    dwp_zero128<<<(unsigned)((nf4 + TB - 1) / TB), TB, 0, stream>>>((float4*)outFeats, nf4);
  }
  // 1-2) presence masks
  dwp_mark_cand<<<(M + TB - 1) / TB, TB, 0, stream>>>(xc, candMask, M);
  dwp_mark_ref<<<(NREF + TB - 1) / TB, TB, 0, stream>>>(rc, refMask, NREF);
  // 3-5) hierarchical popcount prefix over the 2^24-bit mask
  dwp_scan_tiles<<<NTILES, 256, 0, stream>>>(candMask, wordPrefix, tileSums);
  dwp_scan_tilesums<<<1, NTILES, 0, stream>>>(tileSums);
  dwp_add_offsets<<<NWORDS / TB, TB, 0, stream>>>(wordPrefix, tileSums);
  // 6-7) coords + keep outputs
  dwp_fill_defaults<<<(M + TB - 1) / TB, TB, 0, stream>>>(xc, outCoords, outKeep, M);
  dwp_write_unique<<<NWORDS / TB, TB, 0, stream>>>(candMask, wordPrefix, refMask,
                                                   outCoords, outKeep);
  // 8) WMMA GEMM fused with pruned dedup scatter
  int rowTiles = N / 16;
  if (rowTiles > 0)
    dwp_gemm_scatter<<<rowTiles, 128, 0, stream>>>(xf, xc, W, candMask, wordPrefix,
                                                   refMask, outFeats);
  int rem = N - rowTiles * 16;
  if (rem > 0) {
    int work = rem * NKERN * COUT;
    dwp_gemm_tail<<<(work + TB - 1) / TB, TB, 0, stream>>>(xf, xc, W, candMask,
                                                           wordPrefix, refMask,
                                                           outFeats, rowTiles * 16, rem);
  }
  // 9) bias for kept rows (non-kept rows remain zero from the init pass)
  size_t total = (size_t)M * COUT;
  dwp_finalize<<<(unsigned)((total + TB - 1) / TB), TB, 0, stream>>>(outFeats, outKeep,
                                                                     bias, total);
}